// SmithWatermanLoss_62551903699009
// MI455X (gfx1250) — compile-verified
//
#include <hip/hip_runtime.h>
#include <cstdint>

#define NEG_INF   (-1.0e30f)
#define GAP_OPEN  (-5.0f)
#define GAP_EXT   (-1.0f)

constexpr int NDIM  = 151;   // N = M
constexpr int A     = 150;   // a = b = N-1
constexpr int NDIAG = 2*A - 1; // 299
constexpr int CPL   = 5;     // cells per lane (5*30 = 150 real positions)
constexpr int WAVES = 8;     // waves per block (one batch per wave)
constexpr int BUFW  = 160;   // padded diagonal buffer (32 lanes * 5)

// ---- CDNA5 async global->LDS DMA (ASYNCcnt path) -------------------------
__device__ __forceinline__ void async_ld_f32(unsigned lds_off, const float* gaddr) {
  asm volatile("global_load_async_to_lds_b32 %0, %1, off"
               :: "v"(lds_off), "v"(gaddr)
               : "memory");
}
__device__ __forceinline__ void wait_async_le(int) {}
__device__ __forceinline__ void wait_async_5() {
  asm volatile("s_wait_asynccnt 5" ::: "memory");
}
__device__ __forceinline__ void wait_async_0() {
  asm volatile("s_wait_asynccnt 0" ::: "memory");
}

// ---- fast LSE helpers (hardware v_exp_f32 / v_log_f32) -------------------
__device__ __forceinline__ float lse2(float a, float b) {
  float m = fmaxf(a, b);
  return m + __logf(__expf(a - m) + __expf(b - m));
}
__device__ __forceinline__ float lse3(float a, float b, float c) {
  float m = fmaxf(fmaxf(a, b), c);
  return m + __logf(__expf(a - m) + __expf(b - m) + __expf(c - m));
}
__device__ __forceinline__ float lse4(float a, float b, float c, float d) {
  float m = fmaxf(fmaxf(a, b), fmaxf(c, d));
  return m + __logf(__expf(a - m) + __expf(b - m) + __expf(c - m) + __expf(d - m));
}

__global__ __launch_bounds__(WAVES * 32)
void sw_forward_kernel(const float* __restrict__ x, float* __restrict__ bout, int B) {
  __shared__ float diagbuf[WAVES][2][BUFW];

  const int lane = threadIdx.x & 31;
  const int wv   = threadIdx.x >> 5;
  const int b    = blockIdx.x * WAVES + wv;
  if (b >= B) return;                       // wave-uniform exit

  const float* xb = x + (size_t)b * (NDIM * NDIM);

  float h1[CPL][3], h2[CPL][3];
#pragma unroll
  for (int e = 0; e < CPL; ++e)
#pragma unroll
    for (int k = 0; k < 3; ++k) { h1[e][k] = NEG_INF; h2[e][k] = NEG_INF; }

  // online logsumexp accumulator (per lane)
  float mx = -3.0e38f, s = 0.0f;

  // ---- prefetch diagonal 0 into buffer 0 ----
  {
    const int t = 0;
    const int p = (t + 1) & 1;
    const int base = (t + (A - 1) - p) >> 1;
#pragma unroll
    for (int e = 0; e < CPL; ++e) {
      int j = lane * CPL + e;
      int r = base - j, c = t - r;
      bool valid = (r >= 0) & (r < A) & (c >= 0) & (c < A);
      const float* g = valid ? (xb + r * NDIM + c) : xb;   // clamp to in-bounds
      unsigned off = (unsigned)(size_t)&diagbuf[wv][0][j];
      async_ld_f32(off, g);
    }
  }

  for (int t = 0; t < NDIAG; ++t) {
    const int so   = t & 1;                  // smo[t] (a even -> t%2)
    const int p    = (t + 1) & 1;
    const int base = (t + (A - 1) - p) >> 1;

    // ---- issue async DMA for diagonal t+1 into the other buffer ----
    if (t + 1 < NDIAG) {
      const int t2 = t + 1;
      const int p2 = (t2 + 1) & 1;
      const int b2 = (t2 + (A - 1) - p2) >> 1;
#pragma unroll
      for (int e = 0; e < CPL; ++e) {
        int j = lane * CPL + e;
        int r = b2 - j, c = t2 - r;
        bool valid = (r >= 0) & (r < A) & (c >= 0) & (c < A);
        const float* g = valid ? (xb + r * NDIM + c) : xb;
        unsigned off = (unsigned)(size_t)&diagbuf[wv][t2 & 1][j];
        async_ld_f32(off, g);
      }
      wait_async_5();   // 10 outstanding -> wait until only the 5 new remain
    } else {
      wait_async_0();   // last diagonal: drain
    }

    // ---- consume diagonal t from LDS; issue xp (= x[b,r+1,c+1]) loads early
    float sx[CPL], xp[CPL];
    bool  val[CPL];
#pragma unroll
    for (int e = 0; e < CPL; ++e) {
      int j = lane * CPL + e;
      int r = base - j, c = t - r;
      val[e] = (r >= 0) & (r < A) & (c >= 0) & (c < A);
      float v = diagbuf[wv][t & 1][j];
      sx[e] = val[e] ? v : NEG_INF;
      xp[e] = val[e] ? xb[(r + 1) * NDIM + (c + 1)] : 0.0f;
    }

    // ---- cross-lane neighbor exchange (wave32) ----
    float pl[3], nl[3];
#pragma unroll
    for (int k = 0; k < 3; ++k) {
      float u = __shfl_up(h1[CPL - 1][k], 1, 32);
      pl[k] = (lane == 0) ? NEG_INF : u;        // h1[pos-1] boundary
      float d = __shfl_down(h1[0][k], 1, 32);
      nl[k] = (lane == 31) ? NEG_INF : d;       // h1[pos+1] boundary
    }

    // ---- cell updates ----
    float h0[CPL][3];
#pragma unroll
    for (int e = 0; e < CPL; ++e) {
      float hp0 = (e == 0) ? pl[0] : h1[e - 1][0];
      float hp1 = (e == 0) ? pl[1] : h1[e - 1][1];
      float hn0 = (e == CPL - 1) ? nl[0] : h1[e + 1][0];
      float hn1 = (e == CPL - 1) ? nl[1] : h1[e + 1][1];
      float hn2 = (e == CPL - 1) ? nl[2] : h1[e + 1][2];

      float r0 = so ? hp0 : h1[e][0];
      float r1 = so ? hp1 : h1[e][1];
      float d0 = so ? h1[e][0] : hn0;
      float d1 = so ? h1[e][1] : hn1;
      float d2 = so ? h1[e][2] : hn2;

      h0[e][0] = lse4(h2[e][0], h2[e][1], h2[e][2], 0.0f) + sx[e];
      h0[e][1] = lse2(r0 + GAP_OPEN, r1 + GAP_EXT);
      h0[e][2] = lse3(d0 + GAP_OPEN, d1 + GAP_OPEN, d2 + GAP_EXT);
    }

    // ---- fused final-score accumulation: online LSE over h0 + xp ----
    float v[CPL][3];
    float lm = -3.0e38f;
#pragma unroll
    for (int e = 0; e < CPL; ++e)
#pragma unroll
      for (int k = 0; k < 3; ++k) {
        v[e][k] = h0[e][k] + xp[e];
        if (val[e]) lm = fmaxf(lm, v[e][k]);
      }
    float nm = fmaxf(mx, lm);
    float acc = 0.0f;
#pragma unroll
    for (int e = 0; e < CPL; ++e)
#pragma unroll
      for (int k = 0; k < 3; ++k)
        acc += val[e] ? __expf(v[e][k] - nm) : 0.0f;
    s = s * __expf(mx - nm) + acc;
    mx = nm;

    // ---- roll carry: (h2,h1) <- (h1,h0) ----
#pragma unroll
    for (int e = 0; e < CPL; ++e)
#pragma unroll
      for (int k = 0; k < 3; ++k) { h2[e][k] = h1[e][k]; h1[e][k] = h0[e][k]; }
  }

  // ---- wave-level LSE merge ----
#pragma unroll
  for (int off = 16; off > 0; off >>= 1) {
    float om = __shfl_xor(mx, off, 32);
    float os = __shfl_xor(s,  off, 32);
    float nm = fmaxf(mx, om);
    s = s * __expf(mx - nm) + os * __expf(om - nm);
    mx = nm;
  }
  if (lane == 0) bout[b] = mx + __logf(s);
}

// Deterministic final sum over per-batch LSE values (single block, fixed order).
__global__ __launch_bounds__(256)
void reduce_sum_kernel(const float* __restrict__ ws, float* __restrict__ out, int n) {
  __shared__ float sm[256];
  int tid = threadIdx.x;
  float a = 0.0f;
  for (int i = tid; i < n; i += 256) a += ws[i];
  sm[tid] = a;
  __syncthreads();
  for (int st = 128; st > 0; st >>= 1) {
    if (tid < st) sm[tid] += sm[tid + st];
    __syncthreads();
  }
  if (tid == 0) out[0] = sm[0];
}

extern "C" void kernel_launch(void* const* d_in, const int* in_sizes, int n_in,
                              void* d_out, int out_size, void* d_ws, size_t ws_size,
                              hipStream_t stream) {
  const float* x = (const float*)d_in[0];
  float* ws = (float*)d_ws;                 // B floats of scratch
  const int B = in_sizes[0] / (NDIM * NDIM); // 512
  const int blocks = (B + WAVES - 1) / WAVES;

  sw_forward_kernel<<<blocks, WAVES * 32, 0, stream>>>(x, ws, B);
  reduce_sum_kernel<<<1, 256, 0, stream>>>(ws, (float*)d_out, B);
}